// CTCWithAlignment_867583394548
// MI455X (gfx1250) — compile-verified
//
#include <hip/hip_runtime.h>
#include <hip/hip_bf16.h>

#ifndef __has_builtin
#define __has_builtin(x) 0
#endif

#if __has_builtin(__builtin_amdgcn_global_load_async_to_lds_b32)
#define USE_ASYNC_LDS 1
#else
#define USE_ASYNC_LDS 0
#endif

#define NEGV (-1e30f)

typedef __attribute__((address_space(1))) int GlobalInt;  // global
typedef __attribute__((address_space(3))) int LdsInt;     // LDS

namespace {
constexpr int kV    = 256;            // vocab size
constexpr int kS    = 400;            // max target length
constexpr int kL    = 2 * kS + 1;     // 801 extended CTC states
constexpr int kNTH  = 256;            // 8 wave32 waves per block
constexpr int kPAD  = 4;              // front pad so alpha[l] lives at buf[l+4]
constexpr int kLBUF = kL + 16;        // padded alpha buffer size
constexpr int kNBUF = 8;              // LDS row ring slots (8KB)
constexpr int kDPT  = 6;              // async prefetch depth (hides HBM latency)
}

// 3-way log-sum-exp on the fast VALU path: v_max3 + 3x v_exp + 1x v_log.
// Safe with the -1e30 sentinel: exp underflows to 0; all-NEG stays ~-1e30.
__device__ __forceinline__ float logadd3(float a, float b, float c) {
    float m = fmaxf(fmaxf(a, b), c);
    float s = __expf(a - m) + __expf(b - m) + __expf(c - m);
    return m + __logf(s);
}

__device__ __forceinline__ float logadd2(float a, float b) {
    float m = fmaxf(a, b);
    float s = __expf(a - m) + __expf(b - m);
    return m + __logf(s);
}

__device__ __forceinline__ void stage_row(const float* g, float* l, int tid) {
#if USE_ASYNC_LDS
    // one GLOBAL_LOAD_ASYNC_TO_LDS_B32 per wave (32 lanes x 4B); 8 waves
    // cover the 1KB row. Tracked with ASYNCcnt.
    __builtin_amdgcn_global_load_async_to_lds_b32(
        (GlobalInt*)(void*)(g + tid),
        (LdsInt*)(void*)(l + tid),
        0, 0);
#else
    l[tid] = g[tid];
#endif
}

template <int N>
__device__ __forceinline__ void wait_async() {
#if USE_ASYNC_LDS
#if __has_builtin(__builtin_amdgcn_s_wait_asynccnt)
    __builtin_amdgcn_s_wait_asynccnt(N);
#else
    asm volatile("s_wait_asynccnt %0" :: "n"(N) : "memory");
#endif
#endif
}

__global__ void __launch_bounds__(kNTH)
ctc_loss_kernel(const float* __restrict__ logp,    // [B,T,V]
                const int*   __restrict__ targets, // [B,S]
                const int*   __restrict__ ilen,    // [B]
                const int*   __restrict__ tlen,    // [B]
                float*       __restrict__ out,     // [1]
                int T, int B)
{
    __shared__ __align__(16) float rowbuf[kNBUF][kV]; // logp row ring
    __shared__ __align__(16) float bufA[kLBUF];       // alpha ping (shift +4)
    __shared__ __align__(16) float bufB[kLBUF];       // alpha pong

    const int b   = blockIdx.x;
    const int tid = threadIdx.x;
    const float* lp = logp + (size_t)b * (size_t)T * kV;
    const int il = ilen[b];
    const int tl = tlen[b];
    const int Lv = 2 * tl + 1;
    const int Tl = (il < T) ? il : T;   // t >= il is an identity update: stop there

    // This thread owns states l0..l0+3 (l0 even). Even states are BLANK.
    const int  l0  = tid * 4;
    const bool own = (l0 < kL);
    const int* tgt = targets + (size_t)b * kS;

    int  e1 = 1, e3 = 1;          // ext for odd states l0+1, l0+3
    bool skip1 = false, skip3 = false;
    if (own) {
        int i1 = l0 >> 1;         // (l0+1-1)/2
        e1 = tgt[i1];
        skip1 = (l0 + 1 >= 3) ? (e1 != tgt[i1 - 1]) : true;
        if (l0 + 3 < kL) {
            e3 = tgt[i1 + 1];
            skip3 = (e3 != e1);
        }
    }
    const bool v0 = own && (l0     < Lv);
    const bool v1 = own && (l0 + 1 < Lv);
    const bool v2 = own && (l0 + 2 < Lv);
    const bool v3 = own && (l0 + 3 < Lv);

    // Prologue: stage rows 0..kDPT, wait for our portion of row 0.
    for (int r = 0; r <= kDPT && r < Tl; ++r)
        stage_row(lp + (size_t)r * kV, rowbuf[r % kNBUF], tid);
    wait_async<kDPT>();
    if (tid < kPAD) { bufA[tid] = NEGV; bufB[tid] = NEGV; }
    __syncthreads();

    // alpha0 from row 0
    if (own) {
        const float* row0 = rowbuf[0];
        float4 a;
        a.x = (l0 == 0 && v0) ? row0[1]  : NEGV;   // state 0: BLANK
        a.y = (l0 == 0 && v1) ? row0[e1] : NEGV;   // state 1: tgt[0]
        a.z = NEGV;
        a.w = NEGV;
        *(float4*)(bufA + l0 + kPAD) = a;
    }

    const float* cur = bufA;
    float*       nxt = bufB;

    for (int t = 1; t < Tl; ++t) {
        wait_async<kDPT - 1>();  // our portion of row t landed in LDS
        __syncthreads();         // all waves' row t + prev alpha writes visible

        int r = t + kDPT;
        if (r < Tl) stage_row(lp + (size_t)r * kV, rowbuf[r % kNBUF], tid);

        const float* row = rowbuf[t % kNBUF];
        if (own) {
            // alpha[l0..l0+3] and alpha[l0-2..l0-1] from previous step
            float4 a04 = *(const float4*)(cur + l0 + kPAD);
            float2 am  = *(const float2*)(cur + l0 + kPAD - 2);
            float lpB  = row[1];
            float lp1  = row[e1];
            float lp3  = row[e3];

            float4 n;
            n.x = logadd3(a04.x, am.y,  NEGV)                  + lpB;
            n.y = logadd3(a04.y, a04.x, skip1 ? am.y  : NEGV)  + lp1;
            n.z = logadd3(a04.z, a04.y, NEGV)                  + lpB;
            n.w = logadd3(a04.w, a04.z, skip3 ? a04.y : NEGV)  + lp3;
            n.x = v0 ? n.x : NEGV;
            n.y = v1 ? n.y : NEGV;
            n.z = v2 ? n.z : NEGV;
            n.w = v3 ? n.w : NEGV;
            *(float4*)(nxt + l0 + kPAD) = n;
        }
        const float* tc = nxt; nxt = (float*)cur; cur = tc;
    }
    __syncthreads();

    if (tid == 0) {
        int end = 2 * tl;
        float ll   = logadd2(cur[end + kPAD], cur[end + kPAD - 1]);
        float loss = -ll;
        if (!(loss < 1e29f)) loss = 0.0f;   // also catches NaN
        atomicAdd(out, loss / ((float)tl * (float)B));
    }
}

__global__ void zero_out_kernel(float* out) { out[0] = 0.0f; }

extern "C" void kernel_launch(void* const* d_in, const int* in_sizes, int n_in,
                              void* d_out, int out_size, void* d_ws, size_t ws_size,
                              hipStream_t stream) {
    const float* logp    = (const float*)d_in[0];
    const int*   targets = (const int*)d_in[1];
    const int*   ilen    = (const int*)d_in[2];
    const int*   tlen    = (const int*)d_in[3];
    float*       out     = (float*)d_out;

    const int B = in_sizes[2];                 // 64
    const int T = in_sizes[0] / (B * kV);      // 2000

    zero_out_kernel<<<1, 1, 0, stream>>>(out);
    ctc_loss_kernel<<<dim3(B), dim3(kNTH), 0, stream>>>(
        logp, targets, ilen, tlen, out, T, B);
}